// GraphNetWithSAGPooling_56075093017230
// MI455X (gfx1250) — compile-verified
//
#include <hip/hip_runtime.h>
#include <math.h>

#define NN   100352          // total nodes
#define EE   3211264         // edges
#define BB   64              // graphs
#define NPER 1568            // nodes per graph
#define KEEP 1255            // SAGPool keep per graph
#define N2   80320           // BB*KEEP
#define FIN  128
#define HID  256
#define SORTN 2048

typedef float v2f __attribute__((ext_vector_type(2)));
typedef float v8f __attribute__((ext_vector_type(8)));

// ---------------------------------------------------------------------------
// deg[dst] += w  (conv1 degrees, self-loop +1 applied at use site)
__global__ void deg1_k(const int* __restrict__ ei, const float* __restrict__ w,
                       float* __restrict__ deg) {
    int e = blockIdx.x * 256 + threadIdx.x;
    if (e >= EE) return;
    atomicAdd(&deg[ei[EE + e]], w[e]);
}

// ---------------------------------------------------------------------------
// GEMM1: hw[N,16] = x[N,128] @ W1[128,16]  via V_WMMA_F32_16X16X4_F32.
// One wave per 16-row tile; 32 chained WMMAs over K=128.
__global__ void gemm1_k(const float* __restrict__ x, const float* __restrict__ W1,
                        float* __restrict__ hw) {
    int gw   = (blockIdx.x * 256 + threadIdx.x) >> 5;   // M-tile id (grid exact)
    int lane = threadIdx.x & 31;
    if (gw >= NN / 16) return;                          // wave-uniform
    int mrow  = gw * 16 + (lane & 15);
    int khalf = (lane >> 4) * 2;                        // 0 (lo half) or 2 (hi half)
    int col   = lane & 15;
    const float* xr = x + mrow * FIN;
    v8f acc = {};
#pragma unroll
    for (int k = 0; k < 32; ++k) {
        v2f a = *(const v2f*)(xr + 4 * k + khalf);      // A: rows of x, K pair
        v2f b;
        b.x = W1[(4 * k + khalf)     * 16 + col];       // B: rows of W1
        b.y = W1[(4 * k + khalf + 1) * 16 + col];
        acc = __builtin_amdgcn_wmma_f32_16x16x4_f32(false, a, false, b,
                                                    (short)0, acc, false, false);
    }
    int rbase = gw * 16 + ((lane >> 4) ? 8 : 0);        // C/D layout: M=j / M=j+8
#pragma unroll
    for (int j = 0; j < 8; ++j)
        hw[(rbase + j) * 16 + col] = acc[j];
}

// ---------------------------------------------------------------------------
// conv1 edge aggregation: agg1[dst] += norm * hw[src]  (16 ch)
__global__ void conv1_agg_k(const int* __restrict__ ei, const float* __restrict__ w,
                            const float* __restrict__ deg, const float* __restrict__ hw,
                            float* __restrict__ agg) {
    int e = blockIdx.x * 256 + threadIdx.x;
    if (e >= EE) return;
    int s = ei[e], d = ei[EE + e];
    float nrm = rsqrtf(deg[s] + 1.0f) * w[e] * rsqrtf(deg[d] + 1.0f);
    const float4* hs = (const float4*)(hw + s * 16);
    float* ad = agg + d * 16;
#pragma unroll
    for (int q = 0; q < 4; ++q) {
        float4 v = hs[q];
        atomicAdd(ad + 4 * q + 0, nrm * v.x);
        atomicAdd(ad + 4 * q + 1, nrm * v.y);
        atomicAdd(ad + 4 * q + 2, nrm * v.z);
        atomicAdd(ad + 4 * q + 3, nrm * v.w);
    }
}

// ---------------------------------------------------------------------------
// h = relu(agg1 + hw/deg + b1), written in place over agg1
__global__ void conv1_fin_k(float* __restrict__ agg, const float* __restrict__ hw,
                            const float* __restrict__ deg, const float* __restrict__ b1) {
    int i = blockIdx.x * 256 + threadIdx.x;
    if (i >= NN) return;
    float inv = 1.0f / (deg[i] + 1.0f);
#pragma unroll
    for (int c = 0; c < 16; ++c) {
        float v = agg[i * 16 + c] + hw[i * 16 + c] * inv + b1[c];
        agg[i * 16 + c] = v > 0.0f ? v : 0.0f;
    }
}

// ---------------------------------------------------------------------------
// scorer aggregation (GraphConv, raw weights): sagg[dst] += w * h[src]
__global__ void scorer_agg_k(const int* __restrict__ ei, const float* __restrict__ w,
                             const float* __restrict__ h, float* __restrict__ sagg) {
    int e = blockIdx.x * 256 + threadIdx.x;
    if (e >= EE) return;
    int s = ei[e], d = ei[EE + e];
    float we = w[e];
    const float4* hs = (const float4*)(h + s * 16);
    float* ad = sagg + d * 16;
#pragma unroll
    for (int q = 0; q < 4; ++q) {
        float4 v = hs[q];
        atomicAdd(ad + 4 * q + 0, we * v.x);
        atomicAdd(ad + 4 * q + 1, we * v.y);
        atomicAdd(ad + 4 * q + 2, we * v.z);
        atomicAdd(ad + 4 * q + 3, we * v.w);
    }
}

// ---------------------------------------------------------------------------
// score = tanh(sagg·w_rel + b_s + h·w_root)
__global__ void score_k(const float* __restrict__ h, const float* __restrict__ sagg,
                        const float* __restrict__ wrel, const float* __restrict__ wroot,
                        const float* __restrict__ bs, float* __restrict__ score) {
    int i = blockIdx.x * 256 + threadIdx.x;
    if (i >= NN) return;
    float acc = bs[0];
#pragma unroll
    for (int c = 0; c < 16; ++c)
        acc += sagg[i * 16 + c] * wrel[c] + h[i * 16 + c] * wroot[c];
    score[i] = tanhf(acc);
}

// ---------------------------------------------------------------------------
// Per-graph top-K: bitonic sort (desc) of 1568 scores in LDS -> threshold,
// then index-ordered selection with tie quota (matches jax top_k tie rule).
__global__ void topk_k(const float* __restrict__ score, int* __restrict__ nodemap,
                       int* __restrict__ oldof) {
    __shared__ float sv[SORTN];
    int b = blockIdx.x, t = threadIdx.x;
    int gbase = b * NPER;
    for (int i = t; i < SORTN; i += 256)
        sv[i] = (i < NPER) ? score[gbase + i] : -__builtin_inff();
    __syncthreads();
    for (int k = 2; k <= SORTN; k <<= 1)
        for (int j = k >> 1; j > 0; j >>= 1) {
            for (int i = t; i < SORTN; i += 256) {
                int ixj = i ^ j;
                if (ixj > i) {
                    bool up = ((i & k) == 0);           // descending segment
                    float a = sv[i], c = sv[ixj];
                    if (up ? (a < c) : (a > c)) { sv[i] = c; sv[ixj] = a; }
                }
            }
            __syncthreads();
        }
    float thr = sv[KEEP - 1];                           // K-th largest
    for (int i = t; i < NPER; i += 256)
        nodemap[gbase + i] = -1;
    __syncthreads();
    if (t == 0) {
        int c_gt = 0;
        for (int i = 0; i < NPER; ++i) c_gt += (score[gbase + i] > thr) ? 1 : 0;
        int quota = KEEP - c_gt, taken = 0, rank = 0;
        for (int i = 0; i < NPER && rank < KEEP; ++i) {
            float s = score[gbase + i];
            bool keep = false;
            if (s > thr) keep = true;
            else if (s == thr && taken < quota) { keep = true; ++taken; }
            if (keep) {
                int nid = b * KEEP + rank;
                nodemap[gbase + i] = nid;
                oldof[nid] = gbase + i;
                ++rank;
            }
        }
    }
}

// ---------------------------------------------------------------------------
// hp[new] = h[old] * score[old]  (gate kept nodes)
__global__ void gate_k(const float* __restrict__ h, const float* __restrict__ score,
                       const int* __restrict__ oldof, float* __restrict__ hp) {
    int i = blockIdx.x * 256 + threadIdx.x;
    if (i >= N2) return;
    int old = oldof[i];
    float s = score[old];
    const float4* src4 = (const float4*)(h + old * 16);
    float4* dst4 = (float4*)(hp + i * 16);
#pragma unroll
    for (int q = 0; q < 4; ++q) {
        float4 v = src4[q];
        v.x *= s; v.y *= s; v.z *= s; v.w *= s;
        dst4[q] = v;
    }
}

// ---------------------------------------------------------------------------
// conv2 degrees over remapped (filtered) edges
__global__ void deg2_k(const int* __restrict__ ei, const float* __restrict__ w,
                       const int* __restrict__ nodemap, float* __restrict__ deg2) {
    int e = blockIdx.x * 256 + threadIdx.x;
    if (e >= EE) return;
    int s2 = nodemap[ei[e]], d2 = nodemap[ei[EE + e]];
    if (s2 >= 0 && d2 >= 0) atomicAdd(&deg2[d2], w[e]);
}

// ---------------------------------------------------------------------------
// conv2 edge aggregation in 16-dim (GEMM hoisted out by linearity)
__global__ void conv2_agg_k(const int* __restrict__ ei, const float* __restrict__ w,
                            const int* __restrict__ nodemap, const float* __restrict__ deg2,
                            const float* __restrict__ hp, float* __restrict__ agg2) {
    int e = blockIdx.x * 256 + threadIdx.x;
    if (e >= EE) return;
    int s2 = nodemap[ei[e]], d2 = nodemap[ei[EE + e]];
    if (s2 < 0 || d2 < 0) return;
    float nrm = rsqrtf(deg2[s2] + 1.0f) * w[e] * rsqrtf(deg2[d2] + 1.0f);
    const float4* hs = (const float4*)(hp + s2 * 16);
    float* ad = agg2 + d2 * 16;
#pragma unroll
    for (int q = 0; q < 4; ++q) {
        float4 v = hs[q];
        atomicAdd(ad + 4 * q + 0, nrm * v.x);
        atomicAdd(ad + 4 * q + 1, nrm * v.y);
        atomicAdd(ad + 4 * q + 2, nrm * v.z);
        atomicAdd(ad + 4 * q + 3, nrm * v.w);
    }
}

// ---------------------------------------------------------------------------
// self-loop term: agg2 += hp / (deg2+1), in place (agg2 is GEMM2's A)
__global__ void selfloop2_k(float* __restrict__ agg2, const float* __restrict__ hp,
                            const float* __restrict__ deg2) {
    int i = blockIdx.x * 256 + threadIdx.x;
    if (i >= N2) return;
    float inv = 1.0f / (deg2[i] + 1.0f);
#pragma unroll
    for (int c = 0; c < 16; ++c)
        agg2[i * 16 + c] += hp[i * 16 + c] * inv;
}

// ---------------------------------------------------------------------------
// GEMM2 + relu + mean-pool fused:
// h2 = relu(agg2[N2,16] @ W2[16,256] + b2); out[b] += h2/K per graph.
// One wave per (16-node, 16-col) tile; 4 WMMAs over K=16; run-length
// compressed atomics into out (rows of a lane span at most 2 graphs).
__global__ void gemm2_pool_k(const float* __restrict__ t, const float* __restrict__ W2,
                             const float* __restrict__ b2, float* __restrict__ out) {
    int gw   = (blockIdx.x * 256 + threadIdx.x) >> 5;   // tile id (grid exact)
    int lane = threadIdx.x & 31;
    if (gw >= (N2 / 16) * (HID / 16)) return;           // wave-uniform
    int mt = gw >> 4, nt = gw & 15;
    int mrow  = mt * 16 + (lane & 15);
    int khalf = (lane >> 4) * 2;
    int col   = nt * 16 + (lane & 15);
    v8f acc = {};
#pragma unroll
    for (int k = 0; k < 4; ++k) {
        v2f a = *(const v2f*)(t + mrow * 16 + 4 * k + khalf);
        v2f b;
        b.x = W2[(4 * k + khalf)     * HID + col];
        b.y = W2[(4 * k + khalf + 1) * HID + col];
        acc = __builtin_amdgcn_wmma_f32_16x16x4_f32(false, a, false, b,
                                                    (short)0, acc, false, false);
    }
    float bias = b2[col];
    int rbase = mt * 16 + ((lane >> 4) ? 8 : 0);
    int b0 = rbase / KEEP;
    int bL = (rbase + 7) / KEEP;
    float s0 = 0.0f, s1 = 0.0f;
#pragma unroll
    for (int j = 0; j < 8; ++j) {
        float v = acc[j] + bias;
        v = v > 0.0f ? v : 0.0f;
        if ((rbase + j) / KEEP == b0) s0 += v; else s1 += v;
    }
    const float invK = 1.0f / (float)KEEP;
    atomicAdd(&out[b0 * HID + col], s0 * invK);
    if (bL != b0) atomicAdd(&out[bL * HID + col], s1 * invK);
}

// ---------------------------------------------------------------------------
extern "C" void kernel_launch(void* const* d_in, const int* in_sizes, int n_in,
                              void* d_out, int out_size, void* d_ws, size_t ws_size,
                              hipStream_t stream) {
    const float* x     = (const float*)d_in[0];
    const int*   ei    = (const int*)  d_in[1];   // [2,E] row-major
    const float* ew    = (const float*)d_in[2];
    const float* W1    = (const float*)d_in[4];
    const float* b1    = (const float*)d_in[5];
    const float* wrel  = (const float*)d_in[6];
    const float* wroot = (const float*)d_in[7];
    const float* bs    = (const float*)d_in[8];
    const float* W2    = (const float*)d_in[9];
    const float* b2    = (const float*)d_in[10];
    float* out = (float*)d_out;

    // workspace layout (all 64B aligned: counts are multiples of 16 floats)
    float* ws = (float*)d_ws;
    float* hw    = ws;                    // [NN,16]
    float* agg1  = hw   + (size_t)NN*16;  // [NN,16] -> becomes h in place
    float* sagg  = agg1 + (size_t)NN*16;  // [NN,16]
    float* hp    = sagg + (size_t)NN*16;  // [N2,16]
    float* agg2  = hp   + (size_t)N2*16;  // [N2,16] -> + self-loop = GEMM2 A
    float* deg1  = agg2 + (size_t)N2*16;  // [NN]
    float* score = deg1 + (size_t)NN;     // [NN]
    float* deg2  = score+ (size_t)NN;     // [N2]
    int*   nodemap = (int*)(deg2 + (size_t)N2);   // [NN]
    int*   oldof   = nodemap + (size_t)NN;        // [N2]

    const int EB = (EE + 255) / 256;      // edge-parallel blocks
    const int NB = (NN + 255) / 256;
    const int PB = (N2 + 255) / 256;

    // zero accumulators + output (harness poisons, never re-zeros)
    hipMemsetAsync(deg1, 0, sizeof(float) * NN,      stream);
    hipMemsetAsync(agg1, 0, sizeof(float) * NN * 16, stream);
    hipMemsetAsync(sagg, 0, sizeof(float) * NN * 16, stream);
    hipMemsetAsync(deg2, 0, sizeof(float) * N2,      stream);
    hipMemsetAsync(agg2, 0, sizeof(float) * N2 * 16, stream);
    hipMemsetAsync(out,  0, sizeof(float) * BB * HID, stream);

    deg1_k <<<EB, 256, 0, stream>>>(ei, ew, deg1);
    gemm1_k<<<(NN / 16) / 8, 256, 0, stream>>>(x, W1, hw);          // 6272 waves
    conv1_agg_k<<<EB, 256, 0, stream>>>(ei, ew, deg1, hw, agg1);
    conv1_fin_k<<<NB, 256, 0, stream>>>(agg1, hw, deg1, b1);        // agg1 := h
    scorer_agg_k<<<EB, 256, 0, stream>>>(ei, ew, agg1, sagg);
    score_k<<<NB, 256, 0, stream>>>(agg1, sagg, wrel, wroot, bs, score);
    topk_k<<<BB, 256, 0, stream>>>(score, nodemap, oldof);
    gate_k<<<PB, 256, 0, stream>>>(agg1, score, oldof, hp);
    deg2_k<<<EB, 256, 0, stream>>>(ei, ew, nodemap, deg2);
    conv2_agg_k<<<EB, 256, 0, stream>>>(ei, ew, nodemap, deg2, hp, agg2);
    selfloop2_k<<<PB, 256, 0, stream>>>(agg2, hp, deg2);
    gemm2_pool_k<<<((N2 / 16) * (HID / 16)) / 8, 256, 0, stream>>>(agg2, W2, b2, out);
}